// SelfAttention_90288802497300
// MI455X (gfx1250) — compile-verified
//
#include <hip/hip_runtime.h>

#define B_DIM 16
#define N_DIM 2048
#define H_DIM 1024
#define NT    128   // N_DIM / 16 row tiles

typedef __attribute__((ext_vector_type(2))) float v2f;
typedef __attribute__((ext_vector_type(8))) float v8f;

// ---- f32 WMMA 16x16x4 fragment helpers (ISA 7.12.2 layouts) -----------------

// A (16x4, MxK): lanes 0-15 hold row=lane, VGPR0=K+0, VGPR1=K+1;
//                lanes 16-31 hold row=lane-16, VGPR0=K+2, VGPR1=K+3.
__device__ __forceinline__ v2f load_a_f32(const float* __restrict__ A, int lda,
                                          int r0, int k, int lane) {
  int r  = r0 + (lane & 15);
  int kk = k + ((lane >> 4) << 1);
  const float* p = A + (size_t)r * lda + kk;
  return v2f{p[0], p[1]};
}

// B (4x16, KxN) from row-major KxN matrix (row stride ldb):
// VGPR0 = rows K+0 (lanes 0-15) / K+2 (lanes 16-31), VGPR1 = K+1 / K+3.
__device__ __forceinline__ v2f load_b_rowmajor(const float* __restrict__ Bm, int ldb,
                                               int k, int c0, int lane) {
  int c  = c0 + (lane & 15);
  int kk = k + ((lane >> 4) << 1);
  const float* p = Bm + (size_t)kk * ldb + c;
  return v2f{p[0], p[ldb]};
}

// B (4x16) where B[k][c] = T[c][k], T row-major with row stride ldt (B = T^T).
__device__ __forceinline__ v2f load_b_transposed(const float* __restrict__ T, int ldt,
                                                 int k, int c0, int lane) {
  int c  = c0 + (lane & 15);
  int kk = k + ((lane >> 4) << 1);
  const float* p = T + (size_t)c * ldt + kk;
  return v2f{p[0], p[1]};
}

__device__ __forceinline__ v8f wmma_f32(v2f a, v2f b, v8f c) {
  // 8 args: (neg_a, A, neg_b, B, c_mod, C, reuse_a, reuse_b)
  return __builtin_amdgcn_wmma_f32_16x16x4_f32(false, a, false, b, (short)0, c,
                                               false, false);
}

// ---- Kernel 1: q = feats @ W + bias  (M=B*N=32768, N=1024, K=1024) ----------
// One wave computes a 16x32 output tile (2 accumulators, A reused 2x).
__global__ void fc1_kernel(const float* __restrict__ feats,
                           const float* __restrict__ W,
                           const float* __restrict__ bias,
                           float* __restrict__ q) {
  int gid  = blockIdx.x * (blockDim.x >> 5) + (threadIdx.x >> 5);
  int lane = threadIdx.x & 31;
  int tM = gid >> 5;          // 2048 row tiles
  int tN = gid & 31;          // 32 col tiles of width 32
  int r0 = tM * 16;
  int c0 = tN * 32;

  v8f acc0 = {};
  v8f acc1 = {};
  #pragma unroll 4
  for (int k = 0; k < H_DIM; k += 4) {
    v2f a  = load_a_f32(feats, H_DIM, r0, k, lane);
    v2f b0 = load_b_rowmajor(W, H_DIM, k, c0, lane);
    v2f b1 = load_b_rowmajor(W, H_DIM, k, c0 + 16, lane);
    acc0 = wmma_f32(a, b0, acc0);
    acc1 = wmma_f32(a, b1, acc1);
  }

  // C/D layout: VGPR j holds row j (lanes 0-15) and row j+8 (lanes 16-31).
  int row = r0 + ((lane >> 4) << 3);
  int col = c0 + (lane & 15);
  float bz0 = bias[col];
  float bz1 = bias[col + 16];
  #pragma unroll
  for (int j = 0; j < 8; ++j) {
    q[(size_t)(row + j) * H_DIM + col]      = acc0[j] + bz0;
    q[(size_t)(row + j) * H_DIM + col + 16] = acc1[j] + bz1;
  }
}

// ---- Kernel 2: scores row-slab + softmax + partial column sums --------------
// Block = 8 waves. Handles (b, 16 rows n0..n0+15); score slab 16x2048 in LDS.
__global__ void attn_kernel(const float* __restrict__ q,
                            const float* __restrict__ feats,
                            float* __restrict__ partial) {
  int b  = blockIdx.x >> 7;     // / NT
  int nt = blockIdx.x & (NT - 1);
  int n0 = nt * 16;
  const float* Q = q     + ((size_t)b * N_DIM + n0) * H_DIM;  // 16 x 1024
  const float* F = feats + (size_t)b * N_DIM * H_DIM;         // 2048 x 1024

  extern __shared__ float slab[];   // 16 * 2048 f32 = 128 KB
  __shared__ float rowInv[16];

  int wave = threadIdx.x >> 5;
  int lane = threadIdx.x & 31;

  // GEMM phase: wave owns columns [wave*256, wave*256+256), 4 groups of 4 tiles.
  for (int g = 0; g < 4; ++g) {
    int m0 = wave * 256 + g * 64;
    v8f acc[4] = {};
    #pragma unroll 2
    for (int k = 0; k < H_DIM; k += 4) {
      v2f a = load_a_f32(Q, H_DIM, 0, k, lane);
      #pragma unroll
      for (int t = 0; t < 4; ++t) {
        v2f bf = load_b_transposed(F, H_DIM, k, m0 + t * 16, lane);
        acc[t] = wmma_f32(a, bf, acc[t]);
      }
    }
    int rbase = (lane >> 4) << 3;
    int cl    = lane & 15;
    #pragma unroll
    for (int t = 0; t < 4; ++t) {
      #pragma unroll
      for (int j = 0; j < 8; ++j)
        slab[(rbase + j) * N_DIM + m0 + t * 16 + cl] = acc[t][j];
    }
  }
  __syncthreads();

  // Softmax: wave handles rows 2*wave and 2*wave+1.
  for (int rr = 0; rr < 2; ++rr) {
    int r = wave * 2 + rr;
    float mx = -3.402823466e38f;
    for (int m = lane; m < N_DIM; m += 32)
      mx = fmaxf(mx, slab[r * N_DIM + m]);
    #pragma unroll
    for (int o = 16; o > 0; o >>= 1)
      mx = fmaxf(mx, __shfl_xor(mx, o, 32));
    float s = 0.f;
    for (int m = lane; m < N_DIM; m += 32) {
      float e = __expf(slab[r * N_DIM + m] - mx);
      slab[r * N_DIM + m] = e;
      s += e;
    }
    #pragma unroll
    for (int o = 16; o > 0; o >>= 1)
      s += __shfl_xor(s, o, 32);
    if (lane == 0) rowInv[r] = 1.0f / s;
  }
  __syncthreads();

  // Deterministic partial column sums for this 16-row slab.
  float* P = partial + ((size_t)b * NT + nt) * N_DIM;
  for (int m = threadIdx.x; m < N_DIM; m += 256) {
    float a = 0.f;
    #pragma unroll
    for (int r = 0; r < 16; ++r)
      a += slab[r * N_DIM + m] * rowInv[r];
    P[m] = a;
  }
}

// ---- Kernel 3: colsum[b,m] = sum_t partial[b,t,m] ---------------------------
__global__ void colsum_kernel(const float* __restrict__ partial,
                              float* __restrict__ colsum) {
  int idx = blockIdx.x * blockDim.x + threadIdx.x;   // b*2048 + m
  int b = idx >> 11;
  int m = idx & (N_DIM - 1);
  const float* P = partial + (size_t)b * NT * N_DIM + m;
  float a = 0.f;
  #pragma unroll 8
  for (int t = 0; t < NT; ++t) a += P[(size_t)t * N_DIM];
  colsum[idx] = a;
}

// ---- Kernel 4: out[b,h] = sum_m colsum[b,m] * feats[b,m,h] ------------------
__global__ void pool_kernel(const float* __restrict__ colsum,
                            const float* __restrict__ feats,
                            float* __restrict__ out) {
  int b = blockIdx.x;
  int h = blockIdx.y * 256 + threadIdx.x;
  const float* F = feats + (size_t)b * N_DIM * H_DIM + h;
  const float* C = colsum + (size_t)b * N_DIM;
  float a = 0.f;
  #pragma unroll 4
  for (int m = 0; m < N_DIM; ++m)
    a += C[m] * F[(size_t)m * H_DIM];
  out[(size_t)b * H_DIM + h] = a;
}

// ---- Launch -----------------------------------------------------------------
extern "C" void kernel_launch(void* const* d_in, const int* in_sizes, int n_in,
                              void* d_out, int out_size, void* d_ws, size_t ws_size,
                              hipStream_t stream) {
  const float* feats = (const float*)d_in[0];
  const float* W     = (const float*)d_in[1];
  const float* bias  = (const float*)d_in[2];
  float* out = (float*)d_out;

  // Workspace layout: q (134 MB) | partial (16 MB) | colsum (128 KB)
  float* q       = (float*)d_ws;
  float* partial = q + (size_t)B_DIM * N_DIM * H_DIM;
  float* colsum  = partial + (size_t)B_DIM * NT * N_DIM;

  // fc1: 2048 row-tiles * 32 col-tiles = 65536 waves / 8 per block
  fc1_kernel<<<8192, 256, 0, stream>>>(feats, W, bias, q);

  // attention: one block per (b, 16-row tile); 128 KB dynamic LDS slab
  attn_kernel<<<B_DIM * NT, 256, 16 * N_DIM * sizeof(float), stream>>>(q, feats, partial);

  colsum_kernel<<<(B_DIM * N_DIM) / 256, 256, 0, stream>>>(partial, colsum);

  pool_kernel<<<dim3(B_DIM, H_DIM / 256), 256, 0, stream>>>(colsum, feats, out);
}